// NeighbourKNN_81595788689663
// MI455X (gfx1250) — compile-verified
//
#include <hip/hip_runtime.h>
#include <stdint.h>

// KNN (B=4, N=8192, C=16, k=20) for MI455X / gfx1250, wave32.
// Fused distance + top-k: WMMA f32 16x16x4 inner products -> LDS distance
// tiles -> per-lane register sorted top-20 -> wave shuffle merge.
// v2: tile loop has a compile-time trip count (4 tiles/wave, unrolled) so no
// EXEC masking surrounds the WMMAs and the 4 tiles software-pipeline.

typedef float v2f __attribute__((ext_vector_type(2)));
typedef float v8f __attribute__((ext_vector_type(8)));

#define NPTS     8192
#define NCH      16
#define KNN      20
#define ROWS     16      // query rows per workgroup (== one WMMA M tile)
#define CHUNK    1024    // candidate columns per LDS pass
#define NTHREADS 512     // 16 waves of 32
#define NWAVES   16
#define TILES_PER_WAVE (CHUNK / 16 / NWAVES)   // = 4, compile-time constant
#define LSTRIDE  1028    // 1024 + 4 pad: bank = (4*row + col) % 64, conflict-free

__device__ __forceinline__ unsigned ordered_u32(float f) {
  unsigned u = __float_as_uint(f);
  return (u & 0x80000000u) ? ~u : (u | 0x80000000u); // monotone float->uint
}

__global__ __launch_bounds__(NTHREADS, 1)
void knn_wmma_kernel(const float* __restrict__ x, int* __restrict__ idx_out)
{
  __shared__ float q[ROWS][NCH];          // 1 KB   query tile
  __shared__ float qxx[ROWS];             //        query squared norms
  __shared__ float cxx[CHUNK];            // 4 KB   candidate squared norms
  __shared__ float dist[ROWS * LSTRIDE];  // ~64 KB distance tile (padded)

  const int b    = blockIdx.y;
  const int row0 = blockIdx.x * ROWS;
  const int tid  = threadIdx.x;
  const int wave = tid >> 5;
  const int lane = tid & 31;

  const float* __restrict__ xb = x + (size_t)b * NPTS * NCH;

  // stage the 16 query rows + their squared norms
  if (tid < ROWS * NCH) {
    int r = tid >> 4, c = tid & 15;
    q[r][c] = xb[(size_t)(row0 + r) * NCH + c];
  }
  __syncthreads();
  if (tid < ROWS) {
    float s = 0.f;
#pragma unroll
    for (int c = 0; c < NCH; ++c) s += q[tid][c] * q[tid][c];
    qxx[tid] = s;
  }
  __syncthreads();

  // A-fragments (16x4 f32 per K-chunk; ISA layout: lanes 0-15 hold K=0,1;
  // lanes 16-31 hold K=2,3). Loop-invariant: the same 16 query rows always.
  const int m     = lane & 15;
  const int khalf = (lane >> 4) * 2;
  v2f afrag[4];
#pragma unroll
  for (int kc = 0; kc < 4; ++kc) {
    afrag[kc].x = q[m][4 * kc + khalf];
    afrag[kc].y = q[m][4 * kc + khalf + 1];
  }

  // per-lane sorted (ascending) top-20, fully register resident
  float best[KNN]; int bidx[KNN];
#pragma unroll
  for (int p = 0; p < KNN; ++p) { best[p] = __builtin_inff(); bidx[p] = 0x7fffffff; }

  const int moff = (lane >> 4) * 8; // C/D layout: VGPR r -> rows r (lo) / r+8 (hi)

  for (int chunk0 = 0; chunk0 < NPTS; chunk0 += CHUNK) {
    // squared norms of this chunk's candidates (x is L2-resident, 2 MB total)
#pragma unroll
    for (int jj = 0; jj < CHUNK / NTHREADS; ++jj) {      // exactly 2 iterations
      const int j = tid + jj * NTHREADS;
      const float* p = xb + (size_t)(chunk0 + j) * NCH;
      float s = 0.f;
#pragma unroll
      for (int c = 0; c < NCH; ++c) s += p[c] * p[c];
      cxx[j] = s;
    }
    __syncthreads();

    // -------- compute phase: 64 16x16 tiles, 4 per wave, 4 WMMA each --------
    // Fixed trip count -> fully unrolled, no EXEC masking around WMMA,
    // 16 static v_wmma per chunk iteration, loads clause/pipeline freely.
#pragma unroll
    for (int tt = 0; tt < TILES_PER_WAVE; ++tt) {
      const int t    = wave + tt * NWAVES;
      const int lcol = t * 16 + (lane & 15);
      const float* __restrict__ xc = xb + (size_t)(chunk0 + lcol) * NCH;
      v8f acc = {0.f, 0.f, 0.f, 0.f, 0.f, 0.f, 0.f, 0.f};
#pragma unroll
      for (int kc = 0; kc < 4; ++kc) {
        v2f bfrag;
        bfrag.x = xc[4 * kc + khalf];
        bfrag.y = xc[4 * kc + khalf + 1];
        // D = A x B + C   (v_wmma_f32_16x16x4_f32)
        acc = __builtin_amdgcn_wmma_f32_16x16x4_f32(
            false, afrag[kc], false, bfrag, (short)0, acc, false, false);
      }
      const float cn = cxx[lcol];
#pragma unroll
      for (int r = 0; r < 8; ++r) {
        const int row = r + moff;
        dist[row * LSTRIDE + lcol] = qxx[row] - 2.0f * acc[r] + cn;
      }
    }
    __syncthreads();

    // -------- select phase: wave w scans row w of the chunk --------
#pragma unroll 4
    for (int i = 0; i < CHUNK / 32; ++i) {
      const int lcol = lane + 32 * i;
      const float d  = dist[wave * LSTRIDE + lcol];
      const int gcol = chunk0 + lcol;
      if (d < best[KNN - 1]) {
        // unrolled conditional insertion shift (keeps arrays in VGPRs)
        float pb = d; int pi = gcol;
#pragma unroll
        for (int p = 0; p < KNN; ++p) {
          if (d < best[p]) {
            float tb = best[p]; int ti = bidx[p];
            best[p] = pb; bidx[p] = pi;
            pb = tb;  pi = ti;
          }
        }
      }
    }
    __syncthreads();
  }

  // -------- merge: 32 sorted lane lists -> global top-20 of row `wave` -----
  const int grow = row0 + wave;
  int* __restrict__ orow = idx_out + (size_t)(b * NPTS + grow) * KNN;
  for (int round = 0; round < KNN; ++round) {
    const unsigned hkey = ordered_u32(best[0]);
    const unsigned lkey = (unsigned)bidx[0];
    unsigned rh = hkey, rl = lkey;
#pragma unroll
    for (int off = 16; off > 0; off >>= 1) {   // lexicographic (dist, idx) min
      unsigned oh = __shfl_xor(rh, off, 32);
      unsigned ol = __shfl_xor(rl, off, 32);
      if (oh < rh || (oh == rh && ol < rl)) { rh = oh; rl = ol; }
    }
    if (lane == 0) orow[round] = (int)rl;
    if (rh == hkey && rl == lkey) {            // winning lane pops its head
#pragma unroll
      for (int p = 0; p < KNN - 1; ++p) { best[p] = best[p + 1]; bidx[p] = bidx[p + 1]; }
      best[KNN - 1] = __builtin_inff();
      bidx[KNN - 1] = 0x7fffffff;
    }
  }
}

__global__ void copy_x_kernel(const float* __restrict__ x,
                              float* __restrict__ out, int n) {
  int i = blockIdx.x * blockDim.x + threadIdx.x;
  if (i < n) out[i] = x[i];
}

extern "C" void kernel_launch(void* const* d_in, const int* in_sizes, int n_in,
                              void* d_out, int out_size, void* d_ws, size_t ws_size,
                              hipStream_t stream) {
  (void)in_sizes; (void)n_in; (void)d_ws; (void)ws_size; (void)out_size;
  const float* x = (const float*)d_in[0];

  const int x_elems = 4 * NPTS * NCH;          // 524288 floats, output slot 0
  float* outx  = (float*)d_out;
  int*   outix = (int*)d_out + x_elems;        // idx int32 bits, output slot 1

  copy_x_kernel<<<(x_elems + 255) / 256, 256, 0, stream>>>(x, outx, x_elems);
  knn_wmma_kernel<<<dim3(NPTS / ROWS, 4), NTHREADS, 0, stream>>>(x, outix);
}